// PositionalEncoding2D_40364102648389
// MI455X (gfx1250) — compile-verified
//
#include <hip/hip_runtime.h>
#include <hip/hip_bf16.h>

// PositionalEncoding2D: out[b,c,h,w], b=8, c(hd)=256, h=128, w=128, fp32.
// Pure store-bandwidth kernel (input x is shape-only). Values:
//   k = (c & 127) >> 1, invdim = 10000^(-k/128)
//   c < 128 : v = (c odd ? cos : sin)(h * invdim)   -> constant over w, b
//   c >= 128: v = (c odd ? cos : sin)(w * invdim)   -> constant over h, b
// Each thread computes <=4 transcendental values once, then streams
// non-temporal 128-bit stores (wave32-coalesced: lane l covers w=4l..4l+3).

#define B_   8
#define C_   256
#define H_   128
#define W_   128
#define BSTRIDE_F4 ((size_t)(C_ * H_ * W_) / 4)   // float4 elems per batch image

// Native clang vector so __builtin_nontemporal_store accepts it
// (HIP's float4 is a class type and is rejected by the builtin).
typedef float v4f __attribute__((ext_vector_type(4)));

__global__ __launch_bounds__(256)
void posenc2d_store_kernel(float* __restrict__ out) {
    const int c    = blockIdx.x;          // 0..255  (channel)
    const int htl  = blockIdx.y;          // 0..3    (h tile)
    const int tid  = threadIdx.x;         // 0..255
    const int wave = tid >> 5;            // 0..7  (wave32)
    const int lane = tid & 31;            // 0..31

    const int  k       = (c & 127) >> 1;
    const bool use_cos = (c & 1) != 0;
    // invdim = 10000^(-k/128); one fast pow per thread (negligible)
    const float invdim = __powf(10000.0f, -(float)k * (1.0f / 128.0f));

    v4f* __restrict__ out4 = (v4f*)out;

    if (c < 128) {
        // Value depends on h only: each wave handles rows h = wave + 8*(4*htl + i)
        #pragma unroll
        for (int i = 0; i < 4; ++i) {
            const int h = wave + 8 * (4 * htl + i);
            const float theta = (float)h * invdim;
            const float v = use_cos ? __cosf(theta) : __sinf(theta);
            v4f val;
            val.x = v; val.y = v; val.z = v; val.w = v;
            // v4f index: ((b*C + c)*H + h)*32 + lane  (lane -> w = 4*lane..4*lane+3)
            const size_t base = ((size_t)c * H_ + h) * (W_ / 4) + lane;
            #pragma unroll
            for (int b = 0; b < B_; ++b) {
                __builtin_nontemporal_store(val, &out4[base + (size_t)b * BSTRIDE_F4]);
            }
        }
    } else {
        // Value depends on w only: compute this lane's 4 values ONCE,
        // then replay the same 128-bit payload across all h rows and batches.
        v4f val;
        {
            const float t0 = (float)(4 * lane) * invdim;
            const float d  = invdim;
            if (use_cos) {
                val.x = __cosf(t0);
                val.y = __cosf(t0 + d);
                val.z = __cosf(t0 + 2.0f * d);
                val.w = __cosf(t0 + 3.0f * d);
            } else {
                val.x = __sinf(t0);
                val.y = __sinf(t0 + d);
                val.z = __sinf(t0 + 2.0f * d);
                val.w = __sinf(t0 + 3.0f * d);
            }
        }
        #pragma unroll
        for (int i = 0; i < 4; ++i) {
            const int h = wave + 8 * (4 * htl + i);
            const size_t base = ((size_t)c * H_ + h) * (W_ / 4) + lane;
            #pragma unroll
            for (int b = 0; b < B_; ++b) {
                __builtin_nontemporal_store(val, &out4[base + (size_t)b * BSTRIDE_F4]);
            }
        }
    }
}

extern "C" void kernel_launch(void* const* d_in, const int* in_sizes, int n_in,
                              void* d_out, int out_size, void* d_ws, size_t ws_size,
                              hipStream_t stream) {
    (void)d_in; (void)in_sizes; (void)n_in; (void)d_ws; (void)ws_size; (void)out_size;
    float* out = (float*)d_out;
    // grid: 256 channels x 4 h-tiles; 256 threads = 8 wave32s per block.
    dim3 grid(C_, 4, 1);
    dim3 block(256, 1, 1);
    posenc2d_store_kernel<<<grid, block, 0, stream>>>(out);
}